// MovingNCA_4166118277554
// MI455X (gfx1250) — compile-verified
//
#include <hip/hip_runtime.h>
#include <stdint.h>

#define N_IMG    512
#define N_NEO    510
#define C_ST     96
#define ITERS    50
#define THRESH   0.0007f
#define S_DIM    512          // state rows/cols = N_NEO + 2

typedef float v2f  __attribute__((ext_vector_type(2)));
typedef float v8f  __attribute__((ext_vector_type(8)));
typedef unsigned int u32x4 __attribute__((ext_vector_type(4)));
typedef int   i32x4 __attribute__((ext_vector_type(4)));
typedef int   i32x8 __attribute__((ext_vector_type(8)));

// ---------------------------------------------------------------------------
// Kernel 1: comms = 3x3x96 VALID conv of state, via V_WMMA_F32_16X16X4_F32.
// Block = 128 threads = 4 waves. Wave handles a 16x16 cell tile.
// For each state column y (18 per tile) stream an 18x96 slab into LDS with the
// Tensor Data Mover (double buffered), then 72 WMMA K-steps with a banded B.
// ---------------------------------------------------------------------------
__global__ __launch_bounds__(128) void nca_conv_wmma(
    const float* __restrict__ state,
    const float* __restrict__ comms_w,     // (3,3,96,1)
    float* __restrict__ commsBuf)          // (512,512) scalar per cell
{
    __shared__ float wcB[4][288];                          // rows 0..2 = w[dx][dy][c], row 3 = zeros
    __shared__ __attribute__((aligned(16))) float slab[4][2][18 * 96];

    for (int i = threadIdx.x; i < 4 * 288; i += 128) {
        int row = i / 288, k = i - row * 288;
        float w = 0.f;
        if (row < 3) { int dx = k / 96, c = k - dx * 96; w = comms_w[(dx * 3 + row) * 96 + c]; }
        wcB[row][k] = w;
    }
    __syncthreads();

    const int lane = threadIdx.x & 31;
    const int wv   = __builtin_amdgcn_readfirstlane((int)(threadIdx.x >> 5));
    const int m0   = blockIdx.y * 16;                 // cell-row base (== state row base)
    const int n0   = blockIdx.x * 64 + wv * 16;       // cell-col base

    const int mrow = lane & 15;                       // A: M index / B: N index
    const int khi  = (lane >> 4) << 1;                // K sub-offset {0,2}

    const uint32_t dim1 = (uint32_t)(S_DIM - m0);     // rows left in tensor (OOB rows -> 0)

    auto issue_tdm = [&](int yi, int buf) {
        uint64_t ga = (uint64_t)(uintptr_t)(state + ((size_t)m0 * S_DIM + (size_t)(n0 + yi)) * C_ST);
        uint32_t lds = (uint32_t)(uintptr_t)(&slab[wv][buf][0]);
        u32x4 g0;
        g0[0] = 1u;                                               // count = 1 valid descriptor
        g0[1] = lds;                                              // lds_addr (bytes)
        g0[2] = (uint32_t)ga;                                     // global_addr[31:0]
        g0[3] = (uint32_t)((ga >> 32) & 0x01FFFFFFu) | (2u << 30);// global_addr[56:32] | type=2
        i32x8 g1;
        g1[0] = (int)(2u << 16);                                  // data_size = 4B, no flags
        g1[1] = (int)(96u << 16);                                 // tensor_dim0 = 96 (lo16 @ [31:16])
        g1[2] = (int)((dim1 & 0xFFFFu) << 16);                    // tensor_dim1 lo16 @ [31:16]
        g1[3] = (int)(96u << 16);                                 // tile_dim0 = 96 @ [31:16]
        g1[4] = 18;                                               // tile_dim1 = 18, tile_dim2 = 0
        g1[5] = (int)(S_DIM * C_ST);                              // tensor_dim0_stride = 49152 elems
        g1[6] = 0;
        g1[7] = 0;
        i32x4 z = {0, 0, 0, 0};
#if defined(__clang_major__) && (__clang_major__ >= 23)
        i32x8 z8 = {0, 0, 0, 0, 0, 0, 0, 0};
        __builtin_amdgcn_tensor_load_to_lds(g0, g1, z, z, z8, 0);
#else
        __builtin_amdgcn_tensor_load_to_lds(g0, g1, z, z, 0);
#endif
    };

    v8f acc = {0.f, 0.f, 0.f, 0.f, 0.f, 0.f, 0.f, 0.f};

    issue_tdm(0, 0);
    for (int yi = 0; yi < 18; ++yi) {
        if (yi + 1 < 18) {
            issue_tdm(yi + 1, (yi + 1) & 1);
            __builtin_amdgcn_s_wait_tensorcnt((short)1);   // slab for yi is resident
        } else {
            __builtin_amdgcn_s_wait_tensorcnt((short)0);
        }
        __asm__ __volatile__("" ::: "memory");

        // A row m is the contiguous 288-float window at m*96 of the slab.
        const float* aRow = &slab[wv][yi & 1][mrow * 96 + khi];
        // Banded B: this lane's cell-column n=mrow picks weight row d = yi - n.
        int d = yi - mrow;
        int rowIdx = ((unsigned)d < 3u) ? d : 3;           // row 3 is zeros
        const float* bRow = &wcB[rowIdx][khi];

#pragma unroll 8
        for (int t = 0; t < 72; ++t) {                     // K = 288, 4 per WMMA
            v2f a = *(const v2f*)(aRow + 4 * t);
            v2f b = *(const v2f*)(bRow + 4 * t);
            acc = __builtin_amdgcn_wmma_f32_16x16x4_f32(
                false, a, false, b, (short)0, acc, false, false);
        }
    }

    // D layout: VGPR v, lanes 0-15 -> M=v, lanes 16-31 -> M=v+8 ; N = lane%16
#pragma unroll
    for (int v = 0; v < 8; ++v) {
        int m = m0 + v + ((lane >> 4) << 3);
        int n = n0 + (lane & 15);
        if (m < N_NEO && n < N_NEO) commsBuf[m * 512 + n] = acc[v];
    }
}

// ---------------------------------------------------------------------------
// Kernel 2: one wave per cell: percep gather, 98 sigmoids, state += (coalesced
// 32 lanes x 3 channels), perc coordinate walk.
// ---------------------------------------------------------------------------
__global__ __launch_bounds__(128) void nca_update(
    const float* __restrict__ img,
    const float* __restrict__ comms_b,
    const float* __restrict__ perc_w,
    const float* __restrict__ perc_b,
    const float* __restrict__ out_w,       // (2,98)
    const float* __restrict__ out_b,       // (98)
    const float* __restrict__ commsBuf,
    float* __restrict__ state,
    int* __restrict__ perc)
{
    const int lane = threadIdx.x & 31;
    const int cell = blockIdx.x * 4 + (threadIdx.x >> 5);
    if (cell >= N_NEO * N_NEO) return;
    const int r = cell / N_NEO, c = cell - r * N_NEO;

    const int px = perc[2 * cell + 0];
    const int py = perc[2 * cell + 1];
    float percep = perc_b[0];
#pragma unroll
    for (int t = 0; t < 9; ++t)
        percep += img[(px + t / 3) * N_IMG + (py + t % 3)] * perc_w[t];

    const float comms = commsBuf[r * 512 + c] + comms_b[0];

    float* sp = state + ((size_t)(r + 1) * S_DIM + (c + 1)) * C_ST;
#pragma unroll
    for (int q = 0; q < 3; ++q) {
        int k = lane + 32 * q;                             // 96 = 3*32 exactly
        float o = comms * out_w[k] + percep * out_w[98 + k] + out_b[k];
        float s = 1.f / (1.f + __expf(-o));
        sp[k] += s;
    }
    if (lane < 2) {                                        // action channels 96, 97
        int k = 96 + lane;
        float o = comms * out_w[k] + percep * out_w[98 + k] + out_b[k];
        float a = 1.f / (1.f + __expf(-o));
        int delta = (a < -THRESH) ? -1 : ((a > THRESH) ? 1 : 0);
        int p = ((lane == 0) ? px : py) + delta;
        p = p < 0 ? 0 : (p > N_NEO - 1 ? N_NEO - 1 : p);
        perc[2 * cell + lane] = p;
    }
}

__global__ void nca_init(float* __restrict__ state, int* __restrict__ perc) {
    size_t idx = (size_t)blockIdx.x * blockDim.x + threadIdx.x;
    const size_t nstate = (size_t)S_DIM * S_DIM * C_ST;
    if (idx < nstate) state[idx] = 0.f;
    if (idx < (size_t)N_NEO * N_NEO) {
        int r = (int)(idx / N_NEO), c = (int)(idx % N_NEO);
        perc[2 * idx + 0] = r;                             // base[j] == j for these dims
        perc[2 * idx + 1] = c;
    }
}

__global__ void nca_extract(const float* __restrict__ state, float* __restrict__ out) {
    size_t idx = (size_t)blockIdx.x * blockDim.x + threadIdx.x;
    if (idx >= (size_t)N_NEO * N_NEO * 32) return;
    int k = (int)(idx & 31);
    size_t rc = idx >> 5;
    int c = (int)(rc % N_NEO), r = (int)(rc / N_NEO);
    out[idx] = state[((size_t)(r + 1) * S_DIM + (c + 1)) * C_ST + 64 + k];
}

// ---------------------------------------------------------------------------
extern "C" void kernel_launch(void* const* d_in, const int* in_sizes, int n_in,
                              void* d_out, int out_size, void* d_ws, size_t ws_size,
                              hipStream_t stream) {
    (void)in_sizes; (void)n_in; (void)out_size; (void)ws_size;

    const float* img     = (const float*)d_in[0];
    const float* comms_w = (const float*)d_in[1];
    const float* comms_b = (const float*)d_in[2];
    const float* perc_w  = (const float*)d_in[3];
    const float* perc_b  = (const float*)d_in[4];
    const float* out_w   = (const float*)d_in[5];
    const float* out_b   = (const float*)d_in[6];

    // ws layout: state (512*512*96 f32) | commsBuf (512*512 f32) | perc (510*510*2 i32)
    char*  ws       = (char*)d_ws;
    float* state    = (float*)ws;
    float* commsBuf = (float*)(ws + (size_t)S_DIM * S_DIM * C_ST * 4);
    int*   perc     = (int*)(ws + (size_t)S_DIM * S_DIM * C_ST * 4 + (size_t)S_DIM * S_DIM * 4);

    const size_t nstate = (size_t)S_DIM * S_DIM * C_ST;
    nca_init<<<(unsigned)((nstate + 255) / 256), 256, 0, stream>>>(state, perc);

    dim3 cgrid(8, 32);                                     // 8*64 cols x 32*16 rows
    const int ncell = N_NEO * N_NEO;
    const int ublocks = (ncell + 3) / 4;
    for (int it = 0; it < ITERS; ++it) {
        nca_conv_wmma<<<cgrid, 128, 0, stream>>>(state, comms_w, commsBuf);
        nca_update<<<ublocks, 128, 0, stream>>>(img, comms_b, perc_w, perc_b,
                                                out_w, out_b, commsBuf, state, perc);
    }

    const size_t nout = (size_t)N_NEO * N_NEO * 32;
    nca_extract<<<(unsigned)((nout + 255) / 256), 256, 0, stream>>>(state, (float*)d_out);
}